// GCN_ens_2491081032173
// MI455X (gfx1250) — compile-verified
//
#include <hip/hip_runtime.h>
#include <hip/hip_bf16.h>

typedef __attribute__((ext_vector_type(2))) float v2f;
typedef __attribute__((ext_vector_type(8))) float v8f;

#define NNODES 100000
#define NEDGES 1600000
#define INDIM  128
#define HID    32
#define NCLS   40
#define NBR    4

// ---------------- utility kernels ----------------

__global__ void fill_zero_kernel(float* __restrict__ p, int n) {
    int i = blockIdx.x * blockDim.x + threadIdx.x;
    if (i < n) p[i] = 0.0f;
}

__global__ void deg_kernel(const int* __restrict__ src, const int* __restrict__ dst,
                           float* __restrict__ deg_out, float* __restrict__ deg_in, int E) {
    int i = blockIdx.x * blockDim.x + threadIdx.x;
    if (i < E) {
        unsafeAtomicAdd(&deg_out[src[i]], 1.0f);
        unsafeAtomicAdd(&deg_in[dst[i]],  1.0f);
    }
}

__global__ void norm_kernel(float* __restrict__ deg, int n) {
    int i = blockIdx.x * blockDim.x + threadIdx.x;
    if (i < n) deg[i] = rsqrtf(fmaxf(deg[i], 1.0f));
}

// ---------------- WMMA GEMM: out[N, NREAL] = (A[N,K] * rowscale) @ W[K,NREAL] ----------------
// Block covers 64 rows x NPAD cols. Waves arranged 4 row-tiles x (NPAD/16) col-tiles.
// Uses V_WMMA_F32_16X16X4_F32 (full f32 precision; problem is bandwidth-bound anyway).

template<int K, int NPAD, int NREAL>
__global__ void gemm_wmma_kernel(const float* __restrict__ A,
                                 const float* __restrict__ rowscale,
                                 const float* __restrict__ W,
                                 float* __restrict__ out, int nrows) {
    constexpr int CT = NPAD / 16;          // col tiles
    constexpr int NT = 32 * 4 * CT;        // threads per block (4 row-tiles)
    __shared__ float As[64][K + 1];        // +1 pad vs 64-bank conflicts on column reads
    __shared__ float Ws[K][NPAD];

    const int row0 = blockIdx.x * 64;
    const int tid  = threadIdx.x;

    // Stage A (scaled by rowscale) into LDS, zero-pad out-of-range rows.
    for (int i = tid; i < 64 * K; i += NT) {
        int r = i / K, c = i % K;
        int gr = row0 + r;
        float v = 0.0f;
        if (gr < nrows) v = A[(size_t)gr * K + c] * rowscale[gr];
        As[r][c] = v;
    }
    // Stage W into LDS, zero-pad cols NREAL..NPAD-1.
    for (int i = tid; i < K * NPAD; i += NT) {
        int r = i / NPAD, c = i % NPAD;
        Ws[r][c] = (c < NREAL) ? W[r * NREAL + c] : 0.0f;
    }
    __syncthreads();

    const int w    = tid >> 5;
    const int lane = tid & 31;
    const int rt   = w / CT;               // row tile 0..3
    const int ct   = w % CT;               // col tile 0..CT-1
    const int lr   = lane & 15;
    const bool hi  = lane >= 16;

    v8f acc = {};
    for (int kk = 0; kk < K; kk += 4) {
        // A fragment (16x4 f32): v0 = K0 (lanes 0-15) / K2 (lanes 16-31), v1 = K1 / K3
        const int ka = kk + (hi ? 2 : 0);
        v2f a, b;
        a.x = As[rt * 16 + lr][ka];
        a.y = As[rt * 16 + lr][ka + 1];
        // B fragment (4x16 f32): same K split, N striped across lanes
        b.x = Ws[ka][ct * 16 + lr];
        b.y = Ws[ka + 1][ct * 16 + lr];
        acc = __builtin_amdgcn_wmma_f32_16x16x4_f32(
            /*neg_a=*/false, a, /*neg_b=*/false, b,
            /*c_mod=*/(short)0, acc, /*reuse_a=*/false, /*reuse_b=*/false);
    }

    // D layout: VGPR r -> row r (lanes 0-15) / row r+8 (lanes 16-31)
    #pragma unroll
    for (int r = 0; r < 8; r++) {
        int row = row0 + rt * 16 + r + (hi ? 8 : 0);
        int col = ct * 16 + lr;
        if (row < nrows && col < NREAL)
            out[(size_t)row * NREAL + col] = acc[r];
    }
}

// ---------------- edge scatter: agg[dst[e]*DIM+c] += t[src[e]*DIM+c] ----------------

template<int DIM>
__global__ void scatter_kernel(const float* __restrict__ t, const int* __restrict__ src,
                               const int* __restrict__ dst, float* __restrict__ agg, int E) {
    int idx = blockIdx.x * blockDim.x + threadIdx.x;   // < E*DIM <= 64M, fits int
    if (idx >= E * DIM) return;
    int e = idx / DIM;
    int c = idx - e * DIM;
    unsafeAtomicAdd(&agg[dst[e] * DIM + c], t[src[e] * DIM + c]);
}

// ---------------- finalize: f = agg*ndst + bias; h = relu(f) ----------------

__global__ void finalize_kernel(const float* __restrict__ agg, const float* __restrict__ ndst,
                                const float* __restrict__ bias, float* __restrict__ fout,
                                float* __restrict__ hout, int n, int dim) {
    int i = blockIdx.x * blockDim.x + threadIdx.x;
    if (i >= n * dim) return;
    int r = i / dim, c = i - r * dim;
    float v = agg[i] * ndst[r] + bias[c];
    fout[i] = v;
    if (hout) hout[i] = fmaxf(v, 0.0f);
}

// ---------------- launcher ----------------

extern "C" void kernel_launch(void* const* d_in, const int* in_sizes, int n_in,
                              void* d_out, int out_size, void* d_ws, size_t ws_size,
                              hipStream_t stream) {
    const float* x    = (const float*)d_in[0];
    const int*   src  = (const int*)  d_in[1];
    const int*   dst  = (const int*)  d_in[2];
    const float* W0   = (const float*)d_in[3];
    const float* b0   = (const float*)d_in[4];
    const float* W1   = (const float*)d_in[5];
    const float* b1   = (const float*)d_in[6];
    const float* Wout = (const float*)d_in[7];
    const float* bout = (const float*)d_in[8];

    float* out = (float*)d_out;                               // [NBR, N, NCLS]
    float* f0s = out + (size_t)NBR * NNODES * NCLS;           // [NBR, N, HID]
    float* f1s = f0s + (size_t)NBR * NNODES * HID;            // [NBR, N, HID]

    float* ws    = (float*)d_ws;
    float* nsrc  = ws;                 ws += NNODES;           // deg_out^-1/2
    float* ndstv = ws;                 ws += NNODES;           // deg_in^-1/2
    float* t     = ws;                 ws += (size_t)NNODES * NCLS;  // transform buf (max dim 40)
    float* agg   = ws;                 ws += (size_t)NNODES * NCLS;  // scatter accumulator
    float* h     = ws;                 ws += (size_t)NNODES * HID;   // relu'd hidden

    const int TB = 256;
    auto blk = [](long long n, int tb) { return (unsigned)((n + tb - 1) / tb); };

    // degrees -> norms (nsrc,ndstv are contiguous: one fill + one norm pass over 2N)
    fill_zero_kernel<<<blk(2 * NNODES, TB), TB, 0, stream>>>(nsrc, 2 * NNODES);
    deg_kernel<<<blk(NEDGES, TB), TB, 0, stream>>>(src, dst, nsrc, ndstv, NEDGES);
    norm_kernel<<<blk(2 * NNODES, TB), TB, 0, stream>>>(nsrc, 2 * NNODES);

    const unsigned gemm_grid = (NNODES + 63) / 64;

    for (int b = 0; b < NBR; b++) {
        // ---- layer 0: [N,128]@[128,32] ----
        gemm_wmma_kernel<INDIM, HID, HID><<<gemm_grid, 256, 0, stream>>>(
            x, nsrc, W0 + (size_t)b * INDIM * HID, t, NNODES);
        fill_zero_kernel<<<blk((long long)NNODES * HID, TB), TB, 0, stream>>>(agg, NNODES * HID);
        scatter_kernel<HID><<<blk((long long)NEDGES * HID, TB), TB, 0, stream>>>(t, src, dst, agg, NEDGES);
        finalize_kernel<<<blk((long long)NNODES * HID, TB), TB, 0, stream>>>(
            agg, ndstv, b0 + b * HID, f0s + (size_t)b * NNODES * HID, h, NNODES, HID);

        // ---- layer 1: [N,32]@[32,32] ----
        gemm_wmma_kernel<HID, HID, HID><<<gemm_grid, 256, 0, stream>>>(
            h, nsrc, W1 + (size_t)b * HID * HID, t, NNODES);
        fill_zero_kernel<<<blk((long long)NNODES * HID, TB), TB, 0, stream>>>(agg, NNODES * HID);
        scatter_kernel<HID><<<blk((long long)NEDGES * HID, TB), TB, 0, stream>>>(t, src, dst, agg, NEDGES);
        finalize_kernel<<<blk((long long)NNODES * HID, TB), TB, 0, stream>>>(
            agg, ndstv, b1 + b * HID, f1s + (size_t)b * NNODES * HID, h, NNODES, HID);

        // ---- output layer: [N,32]@[32,40] (cols padded to 48 in LDS) ----
        gemm_wmma_kernel<HID, 48, NCLS><<<gemm_grid, 384, 0, stream>>>(
            h, nsrc, Wout + (size_t)b * HID * NCLS, t, NNODES);
        fill_zero_kernel<<<blk((long long)NNODES * NCLS, TB), TB, 0, stream>>>(agg, NNODES * NCLS);
        scatter_kernel<NCLS><<<blk((long long)NEDGES * NCLS, TB), TB, 0, stream>>>(t, src, dst, agg, NEDGES);
        finalize_kernel<<<blk((long long)NNODES * NCLS, TB), TB, 0, stream>>>(
            agg, ndstv, bout + b * NCLS, out + (size_t)b * NNODES * NCLS, nullptr, NNODES, NCLS);
    }
}